// MatchingNetwork_1236950581287
// MI455X (gfx1250) — compile-verified
//
#include <hip/hip_runtime.h>
#include <math.h>

// Problem constants (match reference)
#define NSUP 8192
#define NQRY 8192
#define KDIM 1024   // INPUT_DIM
#define EDIM 512    // EMBED_DIM
#define NCLS 64     // N_CLASSES

typedef __attribute__((ext_vector_type(2))) float v2f;
typedef __attribute__((ext_vector_type(8))) float v8f;

// D(16x16,f32) = A(16x4,f32) x B(4x16,f32) + C   — CDNA5 V_WMMA_F32_16X16X4_F32
__device__ __forceinline__ v8f wmma4f32(v2f a, v2f b, v8f c) {
  return __builtin_amdgcn_wmma_f32_16x16x4_f32(
      /*neg_a=*/false, a, /*neg_b=*/false, b,
      /*c_mod=*/(short)0, c, /*reuse_a=*/false, /*reuse_b=*/false);
}

// Computes acc[4] (wave-private 16x64 slab of X[m0:m0+16,:] @ W) for one
// 256-thread / 8-wave block. Wave w owns columns [64w, 64w+64).
// buf = 16*512 floats of LDS, used as a double-pass A tile (K in two halves).
// WMMA f32 16x16x4 operand layout (ISA 7.12.2):
//   A: lanes 0-15 hold M=lane, VGPR0/1 = K0/K1; lanes 16-31 hold K2/K3.
//   B: lanes 0-15 hold N=lane row K0 (VGPR0) / K1 (VGPR1); lanes 16-31 K2/K3.
__device__ __forceinline__ void embed16(const float* __restrict__ X,
                                        const float* __restrict__ Wg,
                                        float* buf, int m0, int tid,
                                        v8f acc[4]) {
  const int lane  = tid & 31;
  const int wave  = tid >> 5;
  const int mA    = lane & 15;           // A-operand row
  const int khalf = (lane >> 4) << 1;    // 0 for lanes 0-15, 2 for lanes 16-31
  const int ncol  = lane & 15;           // B/D column within tile
  const int nb    = wave << 6;           // wave's N base (0..448)

#pragma unroll
  for (int t = 0; t < 4; ++t)
#pragma unroll
    for (int i = 0; i < 8; ++i) acc[t][i] = 0.0f;

  for (int kh = 0; kh < KDIM; kh += 512) {
    __syncthreads();  // previous-half reads done / block init visible
    // Stage A half-tile 16x512 into LDS, coalesced float4 loads.
    for (int i = tid; i < 2048; i += 256) {
      const int r = i >> 7, c = i & 127;
      ((float4*)buf)[(r << 7) + c] =
          ((const float4*)X)[(size_t)(m0 + r) * (KDIM / 4) + (kh >> 2) + c];
    }
    __syncthreads();
    for (int k0 = 0; k0 < 512; k0 += 4) {
      const v2f a = *(const v2f*)&buf[mA * 512 + k0 + khalf];  // ds_load_b64
      const float* wr = Wg + (size_t)(kh + k0 + khalf) * EDIM + nb + ncol;
#pragma unroll
      for (int t = 0; t < 4; ++t) {
        v2f b;
        b.x = wr[t * 16];         // W[k+khalf  , nb + t*16 + ncol]
        b.y = wr[t * 16 + EDIM];  // W[k+khalf+1, ...]
        acc[t] = wmma4f32(a, b, acc[t]);
      }
    }
  }
}

__global__ void zero_proto_kernel(float* __restrict__ proto) {
  proto[blockIdx.x * 256 + threadIdx.x] = 0.0f;
}

// Support path: embed 16 rows, L2-normalize, scatter-add into class prototypes.
__global__ void __launch_bounds__(256)
support_proto_kernel(const float* __restrict__ X, const int* __restrict__ labels,
                     const float* __restrict__ Wg, float* __restrict__ proto) {
  __shared__ __align__(16) float buf[16 * 512];
  __shared__ float rowss[16];
  __shared__ int lab[16];
  const int tid = threadIdx.x;
  const int m0  = blockIdx.x << 4;
  if (tid < 16) { rowss[tid] = 0.0f; lab[tid] = labels[m0 + tid]; }

  v8f acc[4];
  embed16(X, Wg, buf, m0, tid, acc);

  const int lane  = tid & 31;
  const int wave  = tid >> 5;
  const int rowHi = (lane >> 4) << 3;  // C/D layout: VGPR r -> row r or r+8
  const int ncol  = lane & 15;
  const int nb    = wave << 6;

#pragma unroll
  for (int r = 0; r < 8; ++r) {
    float p = 0.0f;
#pragma unroll
    for (int t = 0; t < 4; ++t) { const float v = acc[t][r]; p += v * v; }
    atomicAdd(&rowss[r + rowHi], p);  // ds float atomic
  }
  __syncthreads();

#pragma unroll
  for (int r = 0; r < 8; ++r) {
    const int row = r + rowHi;
    const float scale = 1.0f / fmaxf(sqrtf(rowss[row]), 1e-12f);
    float* dst = proto + (size_t)lab[row] * EDIM + nb + ncol;
#pragma unroll
    for (int t = 0; t < 4; ++t)
      unsafeAtomicAdd(&dst[t * 16], acc[t][r] * scale);  // global_atomic_add_f32
  }
}

// Query path: embed 16 rows, normalize into LDS, fused scores GEMM vs proto^T
// (split-K over 8 waves), LDS reduce, softmax over 64 classes, write probs.
__global__ void __launch_bounds__(256)
query_scores_kernel(const float* __restrict__ X, const float* __restrict__ Wg,
                    const float* __restrict__ proto, float* __restrict__ out) {
  __shared__ __align__(16) float buf[16 * 512];  // A tile, then q_norm tile
  __shared__ float rowss[16];
  __shared__ float scores[16 * 64];
  const int tid = threadIdx.x;
  const int m0  = blockIdx.x << 4;
  if (tid < 16) rowss[tid] = 0.0f;
  for (int i = tid; i < 16 * 64; i += 256) scores[i] = 0.0f;

  v8f acc[4];
  embed16(X, Wg, buf, m0, tid, acc);

  const int lane  = tid & 31;
  const int wave  = tid >> 5;
  const int rowHi = (lane >> 4) << 3;
  const int ncol  = lane & 15;
  const int nb    = wave << 6;

#pragma unroll
  for (int r = 0; r < 8; ++r) {
    float p = 0.0f;
#pragma unroll
    for (int t = 0; t < 4; ++t) { const float v = acc[t][r]; p += v * v; }
    atomicAdd(&rowss[r + rowHi], p);
  }
  __syncthreads();  // rowss final; all buf reads of embed done

  // q_norm tile into LDS (reuse buf): buf[row][n], row-major 16x512.
#pragma unroll
  for (int r = 0; r < 8; ++r) {
    const int row = r + rowHi;
    const float scale = 1.0f / fmaxf(sqrtf(rowss[row]), 1e-12f);
#pragma unroll
    for (int t = 0; t < 4; ++t)
      buf[row * EDIM + nb + t * 16 + ncol] = acc[t][r] * scale;
  }
  __syncthreads();

  // scores(16x64) = q_norm(16x512) @ proto^T(512x64); split-K across waves:
  // wave -> class tile (wave&3)*16, K half (wave>>2)*256. B[k][n]=proto[n][k]
  // so each lane's (K,K+1) pair is contiguous in memory -> global_load_b64.
  {
    const int c0 = (wave & 3) << 4;
    const int kb = (wave >> 2) << 8;
    const int khalf = (lane >> 4) << 1;
    const int mA = lane & 15;
    v8f s;
#pragma unroll
    for (int i = 0; i < 8; ++i) s[i] = 0.0f;
    const float* pb = proto + (size_t)(c0 + ncol) * EDIM + kb + khalf;
    for (int k0 = 0; k0 < 256; k0 += 4) {
      const v2f a = *(const v2f*)&buf[mA * EDIM + kb + k0 + khalf];
      const v2f b = *(const v2f*)&pb[k0];
      s = wmma4f32(a, b, s);
    }
#pragma unroll
    for (int r = 0; r < 8; ++r)
      atomicAdd(&scores[(r + rowHi) * NCLS + c0 + ncol], s[r]);
  }
  __syncthreads();

  // Softmax over 64 classes, one thread per query row.
  if (tid < 16) {
    float mx = -3.0e38f;
    for (int c = 0; c < NCLS; ++c) mx = fmaxf(mx, scores[tid * NCLS + c]);
    float sum = 0.0f;
    for (int c = 0; c < NCLS; ++c) {
      const float e = __expf(scores[tid * NCLS + c] - mx);
      scores[tid * NCLS + c] = e;
      sum += e;
    }
    const float inv = 1.0f / sum;
    float* o = out + (size_t)(m0 + tid) * NCLS;
    for (int c = 0; c < NCLS; ++c) o[c] = scores[tid * NCLS + c] * inv;
  }
}

extern "C" void kernel_launch(void* const* d_in, const int* in_sizes, int n_in,
                              void* d_out, int out_size, void* d_ws, size_t ws_size,
                              hipStream_t stream) {
  const float* support = (const float*)d_in[0];   // [8192,1024] f32
  const int*   labels  = (const int*)d_in[1];     // [8192] i32
  const float* query   = (const float*)d_in[2];   // [8192,1024] f32
  const float* W       = (const float*)d_in[3];   // [1024,512] f32
  float* out   = (float*)d_out;                   // [8192,64] f32
  float* proto = (float*)d_ws;                    // [64,512] f32 = 128 KB

  zero_proto_kernel<<<(NCLS * EDIM) / 256, 256, 0, stream>>>(proto);
  support_proto_kernel<<<NSUP / 16, 256, 0, stream>>>(support, labels, W, proto);
  query_scores_kernel<<<NQRY / 16, 256, 0, stream>>>(query, W, proto, out);
}